// DfaRnn_76510547411002
// MI455X (gfx1250) — compile-verified
//
#include <hip/hip_runtime.h>
#include <cmath>

typedef __attribute__((ext_vector_type(2))) float v2f;
typedef __attribute__((ext_vector_type(8))) float v8f;
typedef __attribute__((ext_vector_type(4))) int   v4i;

#define Bc 16
#define Tc 2048
#define Dc 512
#define Hc 512
#define LDP 516   // padded LDS row stride (floats): conflict-free b64 fragment loads

#define AS1 __attribute__((address_space(1)))
#define AS3 __attribute__((address_space(3)))

#if defined(__AMDGCN__) && __has_builtin(__builtin_amdgcn_global_load_async_to_lds_b128)
#define USE_ASYNC_LDS 1
#else
#define USE_ASYNC_LDS 0
#endif

// ---------------- Phase 0: zero the global barrier counter ----------------
__global__ void init_counter(unsigned* cnt) {
    if (threadIdx.x == 0 && blockIdx.x == 0) *cnt = 0u;
}

// ---------------- Phase 1: pre = x @ W_xh + b  (in-place into d_out) -----------
// grid = (B*T/64, H/64), block = 128 (4 waves). Each wave: one 16-col N-tile
// across 4 register-blocked M-tiles (64 rows), reusing each B fragment 4x.
__global__ void pre_gemm(const float* __restrict__ x,
                         const float* __restrict__ Wxh,
                         const float* __restrict__ bias,
                         float* __restrict__ out) {
    extern __shared__ float sA[];       // 64 x LDP
    const int tid  = threadIdx.x;
    const int wave = tid >> 5;
    const int lane = tid & 31;
    const int lm   = lane & 15;         // A-row / B-col / D-col index in tile
    const int kb   = (lane >> 4) * 2;   // k sub-offset for A/B fragments
    const int mt   = blockIdx.x;        // 64-row tile of [B*T]
    const int nb   = blockIdx.y;        // 64-col block of H

    // stage 64x512 slice of x into padded LDS (128-bit transfers)
    for (int idx = tid; idx < 64 * (Dc / 4); idx += blockDim.x) {
        int r = idx >> 7, kq = idx & 127;
        *(float4*)&sA[r * LDP + kq * 4] =
            *(const float4*)&x[(size_t)(mt * 64 + r) * Dc + kq * 4];
    }
    __syncthreads();

    const int j = nb * 64 + wave * 16 + lm;   // this lane's output column
    v8f acc0 = {}, acc1 = {}, acc2 = {}, acc3 = {};
    for (int k0 = 0; k0 < Dc; k0 += 4) {
        v2f w;                                 // B fragment, reused by 4 wmmas
        w.x = Wxh[(size_t)(k0 + kb)     * Hc + j];
        w.y = Wxh[(size_t)(k0 + kb + 1) * Hc + j];
        v2f a0 = *(const v2f*)&sA[(0 * 16 + lm) * LDP + k0 + kb];
        v2f a1 = *(const v2f*)&sA[(1 * 16 + lm) * LDP + k0 + kb];
        v2f a2 = *(const v2f*)&sA[(2 * 16 + lm) * LDP + k0 + kb];
        v2f a3 = *(const v2f*)&sA[(3 * 16 + lm) * LDP + k0 + kb];
        acc0 = __builtin_amdgcn_wmma_f32_16x16x4_f32(false, a0, false, w, (short)0, acc0, false, false);
        acc1 = __builtin_amdgcn_wmma_f32_16x16x4_f32(false, a1, false, w, (short)0, acc1, false, false);
        acc2 = __builtin_amdgcn_wmma_f32_16x16x4_f32(false, a2, false, w, (short)0, acc2, false, false);
        acc3 = __builtin_amdgcn_wmma_f32_16x16x4_f32(false, a3, false, w, (short)0, acc3, false, false);
    }
    const float bj = bias[j];
    const int mg0 = mt * 64 + 8 * (lane >> 4);
#pragma unroll
    for (int i = 0; i < 8; ++i) out[(size_t)(mg0 +  0 + i) * Hc + j] = acc0[i] + bj;
#pragma unroll
    for (int i = 0; i < 8; ++i) out[(size_t)(mg0 + 16 + i) * Hc + j] = acc1[i] + bj;
#pragma unroll
    for (int i = 0; i < 8; ++i) out[(size_t)(mg0 + 32 + i) * Hc + j] = acc2[i] + bj;
#pragma unroll
    for (int i = 0; i < 8; ++i) out[(size_t)(mg0 + 48 + i) * Hc + j] = acc3[i] + bj;
}

// ---------------- Phase 2: persistent sequential scan --------------------------
// grid = 16 WGs (co-resident), block = 64 (2 waves). WG g owns columns
// [32g, 32g+32); wave w owns the 16-col WMMA N-tile at 32g + 16w.
// W_hh column slice lives transposed in LDS for the whole scan.
__global__ void rnn_scan(const float* __restrict__ Whh,
                         float* __restrict__ out,
                         unsigned* __restrict__ cnt) {
    extern __shared__ float lds[];
    float* hbuf = lds;                 // 16 x LDP : h_{t-1}
    float* wbuf = lds + 16 * LDP;      // 32 x LDP : W_hh slice, [n_local][k]
    const int tid  = threadIdx.x;
    const int wave = tid >> 5;
    const int lane = tid & 31;
    const int lm   = lane & 15;
    const int kb   = (lane >> 4) * 2;
    const int g    = blockIdx.x;
    const size_t TH = (size_t)Tc * Hc;

    // pin transposed W_hh slice in LDS (once)
    for (int idx = tid; idx < Hc * 32; idx += blockDim.x) {
        int n = idx & 31, k = idx >> 5;
        wbuf[n * LDP + k] = Whh[(size_t)k * Hc + g * 32 + n];
    }
    __syncthreads();

    const int j = g * 32 + wave * 16 + lm;   // this lane's output column

    for (int t = 0; t < Tc; ++t) {
        v8f acc = {};
        if (t > 0) {
            // stage h_{t-1} (16 x 512 fp32 = 32 KB) into LDS
#if USE_ASYNC_LDS
            for (int idx = tid; idx < 16 * (Hc / 4); idx += blockDim.x) {
                int m = idx >> 7, kq = idx & 127;
                const float* gp = out + (size_t)m * TH + (size_t)(t - 1) * Hc + kq * 4;
                __builtin_amdgcn_global_load_async_to_lds_b128(
                    (AS1 v4i*)gp, (AS3 v4i*)&hbuf[m * LDP + kq * 4], 0, 0);
            }
#if __has_builtin(__builtin_amdgcn_s_wait_asynccnt)
            __builtin_amdgcn_s_wait_asynccnt(0);
#else
            asm volatile("s_wait_asynccnt 0" ::: "memory");
#endif
            __syncthreads();
#else
            for (int idx = tid; idx < 16 * (Hc / 4); idx += blockDim.x) {
                int m = idx >> 7, kq = idx & 127;
                *(float4*)&hbuf[m * LDP + kq * 4] =
                    *(const float4*)&out[(size_t)m * TH + (size_t)(t - 1) * Hc + kq * 4];
            }
            __syncthreads();
#endif
            for (int k0 = 0; k0 < Hc; k0 += 4) {
                v2f a = *(const v2f*)&hbuf[lm * LDP + k0 + kb];
                v2f w = *(const v2f*)&wbuf[(wave * 16 + lm) * LDP + k0 + kb];
                acc = __builtin_amdgcn_wmma_f32_16x16x4_f32(false, a, false, w,
                                                            (short)0, acc, false, false);
            }
        }
        // fused epilogue: h_t = tanh(pre_t + h_{t-1} @ W_hh), in place
#pragma unroll
        for (int i = 0; i < 8; ++i) {
            int m = i + 8 * (lane >> 4);                    // batch row
            size_t off = (size_t)m * TH + (size_t)t * Hc + j;
            out[off] = tanhf(out[off] + acc[i]);
        }
        // device-scope grid barrier across the 16 resident WGs
        __threadfence();
        __syncthreads();
        if (tid == 0) {
            __hip_atomic_fetch_add(cnt, 1u, __ATOMIC_RELEASE, __HIP_MEMORY_SCOPE_AGENT);
            while (__hip_atomic_load(cnt, __ATOMIC_ACQUIRE, __HIP_MEMORY_SCOPE_AGENT)
                   < 16u * (unsigned)(t + 1)) {
                __builtin_amdgcn_s_sleep(1);
            }
        }
        __threadfence();
        __syncthreads();
    }
}

extern "C" void kernel_launch(void* const* d_in, const int* in_sizes, int n_in,
                              void* d_out, int out_size, void* d_ws, size_t ws_size,
                              hipStream_t stream) {
    const float* x   = (const float*)d_in[0];
    const float* Wxh = (const float*)d_in[1];
    const float* Whh = (const float*)d_in[2];
    const float* b   = (const float*)d_in[3];
    // d_in[4] = A (DFA feedback matrix): backward-only, unused in forward.
    float*    out = (float*)d_out;
    unsigned* cnt = (unsigned*)d_ws;

    init_counter<<<1, 32, 0, stream>>>(cnt);

    dim3 g1(Bc * Tc / 64, Hc / 64);
    size_t lds1 = (size_t)64 * LDP * sizeof(float);        // 132 KB x-tile
    pre_gemm<<<g1, 128, lds1, stream>>>(x, Wxh, b, out);

    size_t lds2 = (size_t)(16 + 32) * LDP * sizeof(float); // 99 KB h + W_hh slice
    rnn_scan<<<16, 64, lds2, stream>>>(Whh, out, cnt);
}